// OptimizedSampledAttention_58394375356451
// MI455X (gfx1250) — compile-verified
//
#include <hip/hip_runtime.h>
#include <hip/hip_bf16.h>
#include <math.h>

// ---------------------------------------------------------------------------
// OptimizedSampledAttention for MI455X (gfx1250, wave32).
// B=8 H=12 S=4096 D=64 topk=128. Memory-bound (~210MB traffic -> ~9us floor
// at 23.3 TB/s); attention math (0.4 GFLOP) runs on the fp32 matrix pipe via
// V_WMMA_F32_16X16X4_F32 to preserve fp32 precision vs the reference.
// ---------------------------------------------------------------------------

typedef __attribute__((ext_vector_type(2))) float v2f;
typedef __attribute__((ext_vector_type(8))) float v8f;

#define BDIM 8
#define HDIM 12
#define SDIM 4096
#define DDIM 64
#define TOPK 128
#define NBH  (BDIM * HDIM)

// ---------------------------------------------------------------------------
// Kernel 0: zero the 100MB output (reference scatters into zeros).
// ---------------------------------------------------------------------------
__global__ __launch_bounds__(256) void sa_zero_kernel(float4* __restrict__ out, int n4) {
    int i = blockIdx.x * 256 + threadIdx.x;
    if (i < n4) out[i] = make_float4(0.f, 0.f, 0.f, 0.f);
}

// ---------------------------------------------------------------------------
// Kernel 1: importance = mean + std(ddof=1) over D=64, one wave32 per row.
// Each lane loads a float2 -> 256B contiguous per wave.
// ---------------------------------------------------------------------------
__global__ __launch_bounds__(256) void sa_importance_kernel(
    const float* __restrict__ q, float* __restrict__ imp) {
    int lane = threadIdx.x & 31;
    int row  = blockIdx.x * 8 + (threadIdx.x >> 5);   // 8 waves per block
    const float2* qp = (const float2*)(q + (size_t)row * DDIM) + lane;
    float2 ab = *qp;
    float s  = ab.x + ab.y;
    float ss = ab.x * ab.x + ab.y * ab.y;
#pragma unroll
    for (int m = 16; m >= 1; m >>= 1) {
        s  += __shfl_xor(s,  m, 32);
        ss += __shfl_xor(ss, m, 32);
    }
    if (lane == 0) {
        float mean = s * (1.0f / 64.0f);
        float var  = (ss - 64.0f * mean * mean) * (1.0f / 63.0f);
        imp[row] = mean + sqrtf(fmaxf(var, 0.0f));
    }
}

// ---------------------------------------------------------------------------
// Kernel 2: exact top-128 per (B,H) row of 4096 by iterative argmax in LDS.
// Deterministic (smallest index wins ties). ~96 tiny problems; microseconds.
// ---------------------------------------------------------------------------
__global__ __launch_bounds__(256) void sa_topk_kernel(
    const float* __restrict__ imp, int* __restrict__ idxb) {
    __shared__ float vals[SDIM];
    __shared__ float rv[256];
    __shared__ int   ri[256];
    int bh = blockIdx.x, tid = threadIdx.x;
    const float* ip = imp + (size_t)bh * SDIM;
    for (int e = tid; e < SDIM; e += 256) vals[e] = ip[e];
    __syncthreads();
    for (int i = 0; i < TOPK; ++i) {
        float best = -INFINITY; int bi = SDIM;
        for (int e = tid; e < SDIM; e += 256) {
            float vv = vals[e];
            if (vv > best) { best = vv; bi = e; }   // strict > keeps lowest idx
        }
        rv[tid] = best; ri[tid] = bi;
        __syncthreads();
        for (int st = 128; st > 0; st >>= 1) {
            if (tid < st) {
                float ov = rv[tid + st]; int oi = ri[tid + st];
                if (ov > rv[tid] || (ov == rv[tid] && oi < ri[tid])) {
                    rv[tid] = ov; ri[tid] = oi;
                }
            }
            __syncthreads();
        }
        if (tid == 0) { idxb[bh * TOPK + i] = ri[0]; vals[ri[0]] = -INFINITY; }
        __syncthreads();
    }
}

// ---------------------------------------------------------------------------
// Kernel 3: scores = (Qsel * 1/8) @ Ksel^T, row softmax, store P (128x128).
// One block per (B,H), 8 waves; wave w owns rows t = 16w..16w+15.
// A-frag layout (ISA 16x4 f32 A): lane<16 -> K pair {d0,d0+1}, lane>=16 ->
// {d0+2,d0+3}; M = lane%16. B assumed mirrored (N = lane%16, same K split).
// K in LDS padded to 68 floats/row -> conflict-free b64 fragment reads.
// ---------------------------------------------------------------------------
__global__ __launch_bounds__(256) void sa_scores_softmax_kernel(
    const float* __restrict__ q, const float* __restrict__ kmat,
    const int* __restrict__ idxb, float* __restrict__ P) {
    __shared__ float Ks[TOPK][68];
    __shared__ int   sidx[TOPK];
    const int bh = blockIdx.x, tid = threadIdx.x;
    const size_t base = (size_t)bh * SDIM * DDIM;
    if (tid < TOPK) sidx[tid] = idxb[bh * TOPK + tid];
    __syncthreads();
    for (int e = tid; e < TOPK * DDIM; e += 256) {
        int r = e >> 6, col = e & 63;
        Ks[r][col] = kmat[base + (size_t)sidx[r] * DDIM + col];
    }
    __syncthreads();

    const int w = tid >> 5, lane = tid & 31;
    const int n = lane & 15, hl = lane >> 4;
    const int t0 = w * 16;

    // Preload Q A-fragments for this wave's 16-row strip (scale by 1/sqrt(64)).
    v2f afrag[16];
    {
        const float* qp = q + base + (size_t)sidx[t0 + n] * DDIM;
#pragma unroll
        for (int s = 0; s < 16; ++s) {
            int d = 4 * s + 2 * hl;
            afrag[s].x = qp[d]     * 0.125f;
            afrag[s].y = qp[d + 1] * 0.125f;
        }
    }

    v8f acc[8];
#pragma unroll
    for (int j = 0; j < 8; ++j) {
        v8f cacc = {};
        const int s0 = 16 * j;
#pragma unroll
        for (int s = 0; s < 16; ++s) {
            int d = 4 * s + 2 * hl;
            v2f b;
            b.x = Ks[s0 + n][d];
            b.y = Ks[s0 + n][d + 1];
            cacc = __builtin_amdgcn_wmma_f32_16x16x4_f32(
                false, afrag[s], false, b, (short)0, cacc, false, false);
        }
        acc[j] = cacc;
    }

    // Row softmax on accumulator layout: acc[j][r] is row (t0+r+8*hl),
    // column (16j + n). Reduce over j then over the 16 lanes of the half.
#pragma unroll
    for (int r = 0; r < 8; ++r) {
        float m = acc[0][r];
#pragma unroll
        for (int j = 1; j < 8; ++j) m = fmaxf(m, acc[j][r]);
#pragma unroll
        for (int msk = 8; msk >= 1; msk >>= 1) m = fmaxf(m, __shfl_xor(m, msk, 16));
        float sum = 0.f;
#pragma unroll
        for (int j = 0; j < 8; ++j) {
            float p = __expf(acc[j][r] - m);
            acc[j][r] = p;
            sum += p;
        }
#pragma unroll
        for (int msk = 8; msk >= 1; msk >>= 1) sum += __shfl_xor(sum, msk, 16);
        float rinv = 1.0f / sum;
#pragma unroll
        for (int j = 0; j < 8; ++j) acc[j][r] *= rinv;
    }

    float* Pb = P + (size_t)bh * TOPK * TOPK;
#pragma unroll
    for (int r = 0; r < 8; ++r) {
        int t = t0 + r + 8 * hl;
#pragma unroll
        for (int j = 0; j < 8; ++j) Pb[t * TOPK + 16 * j + n] = acc[j][r];
    }
}

// ---------------------------------------------------------------------------
// Kernel 4: O = P @ Vsel (128x128 @ 128x64), scatter rows to out via idx.
// V stored transposed in LDS (Vt[d][s], stride 132) so each B-fragment is a
// conflict-free b64 read of consecutive K (=s) values.
// ---------------------------------------------------------------------------
__global__ __launch_bounds__(256) void sa_pv_scatter_kernel(
    const float* __restrict__ vmat, const float* __restrict__ P,
    const int* __restrict__ idxb, float* __restrict__ out) {
    __shared__ float Vt[DDIM][132];
    __shared__ int   sidx[TOPK];
    const int bh = blockIdx.x, tid = threadIdx.x;
    const size_t base = (size_t)bh * SDIM * DDIM;
    if (tid < TOPK) sidx[tid] = idxb[bh * TOPK + tid];
    __syncthreads();
    for (int e = tid; e < TOPK * DDIM; e += 256) {
        int r = e >> 6, col = e & 63;
        Vt[col][r] = vmat[base + (size_t)sidx[r] * DDIM + col];
    }
    __syncthreads();

    const int w = tid >> 5, lane = tid & 31;
    const int n = lane & 15, hl = lane >> 4;
    const int t0 = w * 16;

    // Preload P A-fragments: 32 K-steps of 4 (K = selected-s dimension, 128).
    v2f af[32];
    {
        const float* Pb = P + (size_t)bh * TOPK * TOPK
                            + (size_t)(t0 + n) * TOPK + 2 * hl;
#pragma unroll
        for (int s = 0; s < 32; ++s) {
            af[s].x = Pb[4 * s];
            af[s].y = Pb[4 * s + 1];
        }
    }

#pragma unroll
    for (int j = 0; j < 4; ++j) {                    // 4 output d-tiles of 16
        const int d0 = 16 * j;
        v8f cacc = {};
#pragma unroll
        for (int s = 0; s < 32; ++s) {
            v2f b;
            b.x = Vt[d0 + n][4 * s + 2 * hl];
            b.y = Vt[d0 + n][4 * s + 2 * hl + 1];
            cacc = __builtin_amdgcn_wmma_f32_16x16x4_f32(
                false, af[s], false, b, (short)0, cacc, false, false);
        }
#pragma unroll
        for (int r = 0; r < 8; ++r) {
            int t = t0 + r + 8 * hl;
            out[base + (size_t)sidx[t] * DDIM + d0 + n] = cacc[r];
        }
    }
}

// ---------------------------------------------------------------------------
// Launch: zero -> importance -> topk -> scores/softmax -> PV/scatter.
// Workspace: imp (1.5MB) | idx (48KB) | P (6MB)  ~= 7.9MB.
// ---------------------------------------------------------------------------
extern "C" void kernel_launch(void* const* d_in, const int* in_sizes, int n_in,
                              void* d_out, int out_size, void* d_ws, size_t ws_size,
                              hipStream_t stream) {
    const float* q = (const float*)d_in[0];
    const float* k = (const float*)d_in[1];
    const float* v = (const float*)d_in[2];
    float* out = (float*)d_out;

    char* ws = (char*)d_ws;
    float* imp  = (float*)ws;                                   // NBH*S floats
    int*   idxb = (int*)(ws + (size_t)NBH * SDIM * 4);          // NBH*TOPK ints
    float* P    = (float*)(ws + (size_t)NBH * SDIM * 4
                              + (size_t)NBH * TOPK * 4);        // NBH*TOPK*TOPK

    int n4 = out_size / 4;
    sa_zero_kernel<<<(n4 + 255) / 256, 256, 0, stream>>>((float4*)out, n4);

    int nrows = NBH * SDIM;
    sa_importance_kernel<<<nrows / 8, 256, 0, stream>>>(q, imp);

    sa_topk_kernel<<<NBH, 256, 0, stream>>>(imp, idxb);

    sa_scores_softmax_kernel<<<NBH, 256, 0, stream>>>(q, k, idxb, P);

    sa_pv_scatter_kernel<<<NBH, 256, 0, stream>>>(v, P, idxb, out);
}